// SegmentorMap_22823456211402
// MI455X (gfx1250) — compile-verified
//
#include <hip/hip_runtime.h>
#include <cstdint>

// ---------------------------------------------------------------------------
// SegmentorMap for MI455X (gfx1250): sparse voxel transformer segmentor.
// - All matmul-shaped work on V_WMMA_F32_16X16X4_F32 (f32 matrix cores,
//   precision-compatible with the f32 reference).
// - GEMM/attention operand tiles staged into LDS by the Tensor Data Mover
//   (tensor_load_to_lds + s_wait_tensorcnt), WMMA fed from ds_load.
// - Pos-bias BatchNorm stats computed analytically from 3x3 covariances
//   (linearity of pos @ pw1), avoiding a 256MB intermediate.
// - SubMConv3d as implicit-GEMM over a dense voxel->row map.
// ---------------------------------------------------------------------------

#define NTOT 4096
#define NSV  256
#define CDIM 128
#define NCLS 20
#define HEADS 8
#define DHD  16
#define EPSV 1e-5f

typedef __attribute__((ext_vector_type(2))) float v2f;
typedef __attribute__((ext_vector_type(8))) float v8f;
typedef __attribute__((ext_vector_type(4))) unsigned int v4u;
typedef __attribute__((ext_vector_type(4))) int v4i;
typedef __attribute__((ext_vector_type(8))) int v8i;

__device__ __forceinline__ v8f wmma_f32(v2f a, v2f b, v8f c) {
  // D(16x16) += A(16x4) * B(4x16), all f32.
  return __builtin_amdgcn_wmma_f32_16x16x4_f32(false, a, false, b, (short)0, c,
                                               false, false);
}

// ------------------- Tensor Data Mover 2D tile staging ---------------------
// Loads a d1 x d0 (rows x contiguous-cols) f32 tile from global (row stride
// `rowstride` elements) into LDS, packed contiguously (row r at lds[r*d0]).
// Descriptor per CDNA5 ISA ch.8: tensor dims == tile dims (tile fully in
// bounds, no clipping needed).
#if __has_builtin(__builtin_amdgcn_tensor_load_to_lds)
#define HAVE_TDM 1
#endif

__device__ __forceinline__ void stage2d(float* lds, const float* g,
                                        int d0, int d1, int rowstride, int lane) {
#if defined(HAVE_TDM)
  (void)lane;
  unsigned long long ga = (unsigned long long)(uintptr_t)g;
  v4u g0;
  g0.x = 1u;                                  // count=1, user descriptor
  g0.y = (unsigned)(uintptr_t)lds;            // LDS byte offset (aperture low bits)
  g0.z = (unsigned)ga;                        // global_addr[31:0]
  g0.w = (unsigned)(ga >> 32) | (2u << 30);   // global_addr[56:32] | type=2
  v8i g1;
  g1[0] = (int)(2u << 16);                    // data_size = 4 bytes
  g1[1] = (int)(((unsigned)d0 & 0xFFFFu) << 16);                      // tensor_dim0 lo
  g1[2] = (int)((((unsigned)d0 >> 16) & 0xFFFFu) |
                (((unsigned)d1 & 0xFFFFu) << 16));                    // td0 hi | td1 lo
  g1[3] = (int)((((unsigned)d1 >> 16) & 0xFFFFu) |
                (((unsigned)d0 & 0xFFFFu) << 16));                    // td1 hi | tile_dim0
  g1[4] = (int)((unsigned)d1 & 0xFFFFu);      // tile_dim1 (tile_dim2 = 0)
  g1[5] = (int)rowstride;                     // tensor_dim0_stride[31:0]
  g1[6] = 0;                                  // stride0 hi | stride1 lo
  g1[7] = 0;
  v4i z4 = {0, 0, 0, 0};
#if defined(__clang_major__) && __clang_major__ >= 23
  v8i z8 = {0, 0, 0, 0, 0, 0, 0, 0};
  __builtin_amdgcn_tensor_load_to_lds(g0, g1, z4, z4, z8, 0);
#else
  __builtin_amdgcn_tensor_load_to_lds(g0, g1, z4, z4, 0);
#endif
#else
  for (int i = lane; i < d0 * d1; i += 32)
    lds[i] = g[(i / d0) * rowstride + (i % d0)];
#endif
}

__device__ __forceinline__ void stage_wait() {
#if defined(HAVE_TDM)
  __builtin_amdgcn_s_wait_tensorcnt(0);
#endif
}

// --------------------------- softmax + segment sum -------------------------
// cf[s,:] = sum over 16 voxels of softmax(features[n,:]) ; segments are
// contiguous groups of 16 rows by construction (super-voxel = n/16).
__global__ void k_softmax_seg(const float* __restrict__ feat,
                              float* __restrict__ cf) {
  __shared__ float red[CDIM];
  int s = blockIdx.x, t = threadIdx.x;
  float acc = 0.f;
  for (int r = 0; r < NTOT / NSV; ++r) {
    float x = feat[(s * 16 + r) * CDIM + t];
    red[t] = x; __syncthreads();
    for (int o = 64; o; o >>= 1) { if (t < o) red[t] = fmaxf(red[t], red[t + o]); __syncthreads(); }
    float mx = red[0]; __syncthreads();
    float e = expf(x - mx);
    red[t] = e; __syncthreads();
    for (int o = 64; o; o >>= 1) { if (t < o) red[t] += red[t + o]; __syncthreads(); }
    float sm = red[0]; __syncthreads();
    acc += e / sm;
  }
  cf[s * CDIM + t] = acc;
}

// --------------------------- WMMA f32 GEMM (TDM-staged) --------------------
// Out[16x16 tile] = act(A) @ W + bias, K = CDIM. One wave per tile. A and W
// tiles are staged to LDS by the TDM; BN+ReLU (if BN) applied once in LDS.
template <int BN>
__global__ void k_gemm(const float* __restrict__ A,
                       const float* __restrict__ W,
                       const float* __restrict__ bias,
                       const float* __restrict__ bn_mu,
                       const float* __restrict__ bn_rs,
                       const float* __restrict__ bn_g,
                       const float* __restrict__ bn_b,
                       float* __restrict__ Out) {
  __shared__ float As[16 * CDIM];   // A tile, row-major [16][K]
  __shared__ float Ws[CDIM * 16];   // W tile, row-major [K][16]
  int lane = threadIdx.x, half = lane >> 4, l = lane & 15;
  int row0 = blockIdx.x * 16, col0 = blockIdx.y * 16;
  stage2d(As, A + (size_t)row0 * CDIM, CDIM, 16, CDIM, lane);
  stage2d(Ws, W + col0, 16, CDIM, CDIM, lane);
  stage_wait();
  __syncthreads();
  if (BN) {
    for (int i = lane; i < 16 * CDIM; i += 32) {
      int k = i & (CDIM - 1);
      float x = As[i];
      As[i] = fmaxf((x - bn_mu[k]) * bn_rs[k] * bn_g[k] + bn_b[k], 0.f);
    }
    __syncthreads();
  }
  v8f acc = {};
#pragma unroll
  for (int k0 = 0; k0 < CDIM; k0 += 4) {
    int ka = k0 + half * 2;          // half-wave owns K pair {ka, ka+1}
    v2f a; a.x = As[l * CDIM + ka]; a.y = As[l * CDIM + ka + 1];
    v2f b; b.x = Ws[ka * 16 + l];   b.y = Ws[(ka + 1) * 16 + l];
    acc = wmma_f32(a, b, acc);
  }
  float bc = bias ? bias[col0 + l] : 0.f;
#pragma unroll
  for (int g = 0; g < 8; ++g)
    Out[(row0 + g + 8 * half) * CDIM + col0 + l] = acc[g] + bc;
}

// --------------------------- BatchNorm statistics --------------------------
__global__ void k_bnstat(const float* __restrict__ x, int M,
                         float* __restrict__ mu, float* __restrict__ rs) {
  int t = threadIdx.x;   // one column per thread (CDIM threads)
  float s = 0.f, s2 = 0.f;
  for (int r = 0; r < M; ++r) { float v = x[r * CDIM + t]; s += v; s2 += v * v; }
  float m = s / (float)M;
  float var = s2 / (float)M - m * m;
  mu[t] = m;
  rs[t] = rsqrtf(var + EPSV);
}

// --------------- analytic BatchNorm stats for the pos-bias MLP -------------
// pos rows are the full cross product a_n - b_s, so
//   mean(h1)_j = (mean(a)-mean(b)) . w_j + pb1_j
//   var(h1)_j  = w_j^T (Cov(a)+Cov(b)) w_j
__global__ void k_posstats(const int* __restrict__ ind,
                           const float* __restrict__ pw1,
                           const float* __restrict__ pb1,
                           float* __restrict__ pmu, float* __restrict__ prs) {
  __shared__ float red[256];
  int t = threadIdx.x;
  float sa0 = 0, sa1 = 0, sa2 = 0;
  float a00 = 0, a01 = 0, a02 = 0, a11 = 0, a12 = 0, a22 = 0;
  for (int n = t; n < NTOT; n += 256) {
    float x = (float)ind[n * 4 + 1], y = (float)ind[n * 4 + 2], z = (float)ind[n * 4 + 3];
    sa0 += x; sa1 += y; sa2 += z;
    a00 += x * x; a01 += x * y; a02 += x * z; a11 += y * y; a12 += y * z; a22 += z * z;
  }
  // one super-voxel row per thread: coords of voxel 16*t, compressed by R=4
  float bx = (float)(ind[t * 64 + 1] >> 2);
  float by = (float)(ind[t * 64 + 2] >> 2);
  float bz = (float)(ind[t * 64 + 3] >> 2);
  float b00 = bx * bx, b01 = bx * by, b02 = bx * bz, b11 = by * by, b12 = by * bz, b22 = bz * bz;
  auto R = [&](float v) -> float {
    red[t] = v; __syncthreads();
    for (int o = 128; o; o >>= 1) { if (t < o) red[t] += red[t + o]; __syncthreads(); }
    float r = red[0]; __syncthreads();
    return r;
  };
  float Sa0 = R(sa0), Sa1 = R(sa1), Sa2 = R(sa2);
  float A00 = R(a00), A01 = R(a01), A02 = R(a02), A11 = R(a11), A12 = R(a12), A22 = R(a22);
  float Sb0 = R(bx), Sb1 = R(by), Sb2 = R(bz);
  float B00 = R(b00), B01 = R(b01), B02 = R(b02), B11 = R(b11), B12 = R(b12), B22 = R(b22);
  const float iN = 1.f / (float)NTOT, iS = 1.f / (float)NSV;
  float ma0 = Sa0 * iN, ma1 = Sa1 * iN, ma2 = Sa2 * iN;
  float mb0 = Sb0 * iS, mb1 = Sb1 * iS, mb2 = Sb2 * iS;
  float Cxx = (A00 * iN - ma0 * ma0) + (B00 * iS - mb0 * mb0);
  float Cxy = (A01 * iN - ma0 * ma1) + (B01 * iS - mb0 * mb1);
  float Cxz = (A02 * iN - ma0 * ma2) + (B02 * iS - mb0 * mb2);
  float Cyy = (A11 * iN - ma1 * ma1) + (B11 * iS - mb1 * mb1);
  float Cyz = (A12 * iN - ma1 * ma2) + (B12 * iS - mb1 * mb2);
  float Czz = (A22 * iN - ma2 * ma2) + (B22 * iS - mb2 * mb2);
  float md0 = ma0 - mb0, md1 = ma1 - mb1, md2 = ma2 - mb2;
  if (t < 64) {
    float w0 = pw1[t], w1 = pw1[64 + t], w2 = pw1[128 + t];
    pmu[t] = md0 * w0 + md1 * w1 + md2 * w2 + pb1[t];
    float var = w0 * w0 * Cxx + w1 * w1 * Cyy + w2 * w2 * Czz +
                2.f * (w0 * w1 * Cxy + w0 * w2 * Cxz + w1 * w2 * Cyz);
    prs[t] = rsqrtf(var + EPSV);
  }
}

// ------------------------------ pos-bias MLP -------------------------------
// One block per voxel n; thread s handles super-voxel s. BN folded to
// scale/shift using analytic stats. pbias[n,s] = relu-MLP(pos) @ pw2 + pb2.
__global__ void k_pbias(const int* __restrict__ ind,
                        const float* __restrict__ pw1, const float* __restrict__ pb1,
                        const float* __restrict__ pg1, const float* __restrict__ pbe1,
                        const float* __restrict__ pw2, const float* __restrict__ pb2,
                        const float* __restrict__ pmu, const float* __restrict__ prs,
                        float* __restrict__ pbias) {
  __shared__ float w[192], sc[64], sh[64], w2[64];
  int t = threadIdx.x;  // 256 threads
  if (t < 64) {
    float g = pg1[t] * prs[t];
    sc[t] = g;
    sh[t] = (pb1[t] - pmu[t]) * g + pbe1[t];
    w2[t] = pw2[t];
  }
  if (t < 192) w[t] = pw1[t];
  __syncthreads();
  int n = blockIdx.x, s = t;
  float ax = (float)ind[n * 4 + 1], ay = (float)ind[n * 4 + 2], az = (float)ind[n * 4 + 3];
  int m0 = s * 64;
  float px = ax - (float)(ind[m0 + 1] >> 2);
  float py = ay - (float)(ind[m0 + 2] >> 2);
  float pz = az - (float)(ind[m0 + 3] >> 2);
  float acc = pb2[0];
#pragma unroll 8
  for (int j = 0; j < 64; ++j) {
    float h1 = px * w[j] + py * w[64 + j] + pz * w[128 + j];
    h1 = fmaxf(h1 * sc[j] + sh[j], 0.f);
    acc = fmaf(h1, w2[j], acc);
  }
  pbias[n * NSV + s] = acc;
}

// --------------------------- fused cross-attention -------------------------
// One wave per (16-query tile, head). q/k/v head slices TDM-staged to LDS;
// scores 16x256 live in LDS; WMMA for both q.k^T and attn.v.
__global__ void k_attn(const float* __restrict__ q, const float* __restrict__ kb,
                       const float* __restrict__ vb, const float* __restrict__ pbias,
                       float* __restrict__ ctx) {
  __shared__ float scn[16][NSV + 4];
  __shared__ float qsh[16 * DHD];
  __shared__ float ksh[NSV * DHD];
  __shared__ float vsh[NSV * DHD];
  int tn = blockIdx.x, h = blockIdx.y;
  int lane = threadIdx.x, half = lane >> 4, l = lane & 15;
  stage2d(qsh, q + (size_t)(tn * 16) * CDIM + h * DHD, DHD, 16, CDIM, lane);
  stage2d(ksh, kb + h * DHD, DHD, NSV, CDIM, lane);
  stage2d(vsh, vb + h * DHD, DHD, NSV, CDIM, lane);
  stage_wait();
  __syncthreads();
  v2f qa[4];
#pragma unroll
  for (int kk = 0; kk < 4; ++kk) {
    int d = kk * 4 + half * 2;
    qa[kk].x = qsh[l * DHD + d];
    qa[kk].y = qsh[l * DHD + d + 1];
  }
  for (int st = 0; st < 16; ++st) {
    v8f acc = {};
    int sr = st * 16 + l;
#pragma unroll
    for (int kk = 0; kk < 4; ++kk) {
      int d = kk * 4 + half * 2;
      v2f b; b.x = ksh[sr * DHD + d]; b.y = ksh[sr * DHD + d + 1];
      acc = wmma_f32(qa[kk], b, acc);
    }
#pragma unroll
    for (int g = 0; g < 8; ++g) {
      int m = g + 8 * half;
      scn[m][st * 16 + l] = acc[g] * 0.25f + pbias[(tn * 16 + m) * NSV + st * 16 + l];
    }
  }
  __syncthreads();
  // row-wise softmax over 256 super-voxels; 8 columns per lane
  for (int r = 0; r < 16; ++r) {
    float mx = -3.0e38f;
#pragma unroll
    for (int j = 0; j < 8; ++j) mx = fmaxf(mx, scn[r][lane * 8 + j]);
    for (int o = 16; o; o >>= 1) mx = fmaxf(mx, __shfl_xor(mx, o, 32));
    float e[8], sm = 0.f;
#pragma unroll
    for (int j = 0; j < 8; ++j) { e[j] = expf(scn[r][lane * 8 + j] - mx); sm += e[j]; }
    for (int o = 16; o; o >>= 1) sm += __shfl_xor(sm, o, 32);
    float inv = 1.f / sm;
#pragma unroll
    for (int j = 0; j < 8; ++j) scn[r][lane * 8 + j] = e[j] * inv;
  }
  __syncthreads();
  v8f cacc = {};
#pragma unroll 4
  for (int s0 = 0; s0 < NSV; s0 += 4) {
    int ks = s0 + half * 2;
    v2f a; a.x = scn[l][ks]; a.y = scn[l][ks + 1];
    v2f b; b.x = vsh[ks * DHD + l]; b.y = vsh[(ks + 1) * DHD + l];
    cacc = wmma_f32(a, b, cacc);
  }
#pragma unroll
  for (int g = 0; g < 8; ++g)
    ctx[(tn * 16 + g + 8 * half) * CDIM + h * DHD + l] = cacc[g];
}

// ---------------------- residual + LayerNorm -------------------------------
__global__ void k_ln(const float* __restrict__ feat, const float* __restrict__ yo,
                     const float* __restrict__ g, const float* __restrict__ b,
                     float* __restrict__ out) {
  __shared__ float red[CDIM];
  int n = blockIdx.x, t = threadIdx.x;
  float y = feat[n * CDIM + t] + yo[n * CDIM + t];
  red[t] = y; __syncthreads();
  for (int o = 64; o; o >>= 1) { if (t < o) red[t] += red[t + o]; __syncthreads(); }
  float mu = red[0] * (1.f / CDIM); __syncthreads();
  float d = y - mu;
  red[t] = d * d; __syncthreads();
  for (int o = 64; o; o >>= 1) { if (t < o) red[t] += red[t + o]; __syncthreads(); }
  float var = red[0] * (1.f / CDIM);
  out[n * CDIM + t] = d * rsqrtf(var + EPSV) * g[t] + b[t];
}

// ---------------------- dense voxel -> row map -----------------------------
__global__ void k_mapinit(int* __restrict__ map) {
  int i = blockIdx.x * blockDim.x + threadIdx.x;
  map[i] = -1;   // grid 2048x256 == 2*128*128*16 exactly
}
__global__ void k_mapset(const int* __restrict__ ind, int* __restrict__ map) {
  int n = blockIdx.x * blockDim.x + threadIdx.x;  // < 4096
  map[((ind[n * 4] * 128 + ind[n * 4 + 1]) * 128 + ind[n * 4 + 2]) * 16 + ind[n * 4 + 3]] = n;
}

// ---------------------- submanifold conv (implicit GEMM) -------------------
// One wave per (16-site tile, 16-wide class tile). Accumulate over the 27
// kernel offsets, gathering neighbor feature rows; empty offsets skipped.
__global__ void k_conv(const float* __restrict__ yln, const int* __restrict__ ind,
                       const int* __restrict__ map, const float* __restrict__ sw,
                       const float* __restrict__ sb, float* __restrict__ out) {
  __shared__ int nid[27][16];
  int tm = blockIdx.x, tnc = blockIdx.y;
  int lane = threadIdx.x, half = lane >> 4, l = lane & 15;
  for (int e = lane; e < 27 * 16; e += 32) {
    int off = e >> 4, m = e & 15;
    int n = tm * 16 + m;
    int bb = ind[n * 4];
    int x = ind[n * 4 + 1] + off / 9 - 1;
    int y = ind[n * 4 + 2] + (off / 3) % 3 - 1;
    int z = ind[n * 4 + 3] + off % 3 - 1;
    int id = -1;
    if (x >= 0 && x < 128 && y >= 0 && y < 128 && z >= 0 && z < 16)
      id = map[((bb * 128 + x) * 128 + y) * 16 + z];
    nid[off][m] = id;
  }
  __syncthreads();
  int nc = tnc * 16 + l;
  int ncc = nc < NCLS ? nc : 0;
  v8f acc = {};
  for (int off = 0; off < 27; ++off) {
    bool any = false;
    for (int m = 0; m < 16; ++m) any = any || (nid[off][m] >= 0);
    if (!any) continue;                       // uniform across the wave
    int arow = nid[off][l];
    int ar = arow < 0 ? 0 : arow;             // safe clamped address
    const float* Wp = sw + off * CDIM * NCLS; // seg_w[kd,kh,kw] block
    for (int k0 = 0; k0 < CDIM; k0 += 4) {
      int ka = k0 + half * 2;
      float a0 = yln[ar * CDIM + ka], a1 = yln[ar * CDIM + ka + 1];
      v2f a; a.x = arow < 0 ? 0.f : a0; a.y = arow < 0 ? 0.f : a1;
      float b0 = Wp[ka * NCLS + ncc], b1 = Wp[(ka + 1) * NCLS + ncc];
      v2f b; b.x = nc < NCLS ? b0 : 0.f; b.y = nc < NCLS ? b1 : 0.f;
      acc = wmma_f32(a, b, acc);
    }
  }
  if (nc < NCLS) {
    float bias = sb[nc];
#pragma unroll
    for (int g = 0; g < 8; ++g)
      out[(tm * 16 + g + 8 * half) * NCLS + nc] = acc[g] + bias;
  }
}

// ---------------------------------------------------------------------------
extern "C" void kernel_launch(void* const* d_in, const int* in_sizes, int n_in,
                              void* d_out, int out_size, void* d_ws, size_t ws_size,
                              hipStream_t stream) {
  (void)in_sizes; (void)n_in; (void)out_size; (void)ws_size;
  const int*   ind  = (const int*)d_in[0];
  const float* feat = (const float*)d_in[1];
  const float* cw1  = (const float*)d_in[2];
  const float* cb1  = (const float*)d_in[3];
  const float* cg1  = (const float*)d_in[4];
  const float* cbe1 = (const float*)d_in[5];
  const float* cw2  = (const float*)d_in[6];
  const float* cb2  = (const float*)d_in[7];
  const float* pw1  = (const float*)d_in[8];
  const float* pb1  = (const float*)d_in[9];
  const float* pg1  = (const float*)d_in[10];
  const float* pbe1 = (const float*)d_in[11];
  const float* pw2  = (const float*)d_in[12];
  const float* pb2  = (const float*)d_in[13];
  const float* wq   = (const float*)d_in[14];
  const float* bq   = (const float*)d_in[15];
  const float* wk   = (const float*)d_in[16];
  const float* bk   = (const float*)d_in[17];
  const float* wv   = (const float*)d_in[18];
  const float* bv   = (const float*)d_in[19];
  const float* wo   = (const float*)d_in[20];
  const float* bo   = (const float*)d_in[21];
  const float* lng  = (const float*)d_in[22];
  const float* lnb  = (const float*)d_in[23];
  const float* sw   = (const float*)d_in[24];
  const float* sb   = (const float*)d_in[25];
  float* out = (float*)d_out;

  float* ws    = (float*)d_ws;
  float* cf    = ws;                 // 256*128
  float* t1    = cf + 32768;         // 256*128
  float* bmu   = t1 + 32768;         // 128
  float* brs   = bmu + 128;          // 128
  float* hbuf  = brs + 128;          // 256*128
  float* qb    = hbuf + 32768;       // 4096*128
  float* kb    = qb + 524288;        // 256*128
  float* vb    = kb + 32768;         // 256*128
  float* pmu   = vb + 32768;         // 64
  float* prs   = pmu + 64;           // 64
  float* pbias = prs + 64;           // 4096*256
  float* ctx   = pbias + 1048576;    // 4096*128
  float* yo    = ctx + 524288;       // 4096*128
  float* yln   = yo + 524288;        // 4096*128
  int*   map   = (int*)(yln + 524288); // 2*128*128*16

  // super-voxel features
  k_softmax_seg<<<NSV, CDIM, 0, stream>>>(feat, cf);
  k_gemm<0><<<dim3(16, 8), 32, 0, stream>>>(cf, cw1, cb1,
      nullptr, nullptr, nullptr, nullptr, t1);
  k_bnstat<<<1, CDIM, 0, stream>>>(t1, NSV, bmu, brs);
  k_gemm<1><<<dim3(16, 8), 32, 0, stream>>>(t1, cw2, cb2,
      bmu, brs, cg1, cbe1, hbuf);
  // projections
  k_gemm<0><<<dim3(256, 8), 32, 0, stream>>>(feat, wq, bq,
      nullptr, nullptr, nullptr, nullptr, qb);
  k_gemm<0><<<dim3(16, 8), 32, 0, stream>>>(hbuf, wk, bk,
      nullptr, nullptr, nullptr, nullptr, kb);
  k_gemm<0><<<dim3(16, 8), 32, 0, stream>>>(hbuf, wv, bv,
      nullptr, nullptr, nullptr, nullptr, vb);
  // positional bias
  k_posstats<<<1, 256, 0, stream>>>(ind, pw1, pb1, pmu, prs);
  k_pbias<<<NTOT, 256, 0, stream>>>(ind, pw1, pb1, pg1, pbe1, pw2, pb2, pmu, prs, pbias);
  // attention + output projection + LayerNorm
  k_attn<<<dim3(NTOT / 16, HEADS), 32, 0, stream>>>(qb, kb, vb, pbias, ctx);
  k_gemm<0><<<dim3(256, 8), 32, 0, stream>>>(ctx, wo, bo,
      nullptr, nullptr, nullptr, nullptr, yo);
  k_ln<<<NTOT, CDIM, 0, stream>>>(feat, yo, lng, lnb, yln);
  // submanifold conv
  k_mapinit<<<2048, 256, 0, stream>>>(map);
  k_mapset<<<16, 256, 0, stream>>>(ind, map);
  k_conv<<<dim3(NTOT / 16, 2), 32, 0, stream>>>(yln, ind, map, sw, sb, out);
}